// SAModule_8761733284506
// MI455X (gfx1250) — compile-verified
//
#include <hip/hip_runtime.h>

#define Bb   16
#define Nn   4096
#define Ss   1024
#define Kk   32
#define CINc 64
#define Mtot (Bb * Ss * Kk)   // 524288 columns
#define R2   (0.2f * 0.2f)

typedef __attribute__((ext_vector_type(2))) float v2f;
typedef __attribute__((ext_vector_type(8))) float v8f;

// ---------------------------------------------------------------------------
// 1) Farthest point sampling: one workgroup per batch, dists cached in LDS,
//    block-wide argmax per step (tie -> lowest index, matching jnp.argmax).
// ---------------------------------------------------------------------------
__global__ __launch_bounds__(512) void fps_kernel(const float* __restrict__ xyz,
                                                  int* __restrict__ fpsIdx,
                                                  float* __restrict__ nxyz,
                                                  float* __restrict__ outXyz) {
  const int b = blockIdx.x;
  const int tid = threadIdx.x;
  __shared__ float dists[Nn];
  __shared__ float rv[512];
  __shared__ int ri[512];
  const float* xb = xyz + (size_t)b * 3 * Nn;
  for (int n = tid; n < Nn; n += 512) dists[n] = 1e10f;
  __syncthreads();
  int last = 0;
  for (int i = 0; i < Ss; ++i) {
    if (tid == 0) {
      fpsIdx[b * Ss + i] = last;
      float cx = xb[last], cy = xb[Nn + last], cz = xb[2 * Nn + last];
      float* nx = nxyz + ((size_t)b * Ss + i) * 3;
      nx[0] = cx; nx[1] = cy; nx[2] = cz;
      outXyz[((size_t)b * 3 + 0) * Ss + i] = cx;
      outXyz[((size_t)b * 3 + 1) * Ss + i] = cy;
      outXyz[((size_t)b * 3 + 2) * Ss + i] = cz;
    }
    if (i == Ss - 1) break;
    const float lx = xb[last], ly = xb[Nn + last], lz = xb[2 * Nn + last];
    float bd = -1.0f; int bi = 0;
    for (int n = tid; n < Nn; n += 512) {
      float dx = xb[n] - lx, dy = xb[Nn + n] - ly, dz = xb[2 * Nn + n] - lz;
      float d = dx * dx + dy * dy + dz * dz;
      float nd = fminf(dists[n], d);
      dists[n] = nd;
      if (nd > bd) { bd = nd; bi = n; }
    }
    rv[tid] = bd; ri[tid] = bi;
    __syncthreads();
    for (int st = 256; st > 0; st >>= 1) {
      if (tid < st) {
        float ov = rv[tid + st]; int oi = ri[tid + st];
        if (ov > rv[tid] || (ov == rv[tid] && oi < ri[tid])) { rv[tid] = ov; ri[tid] = oi; }
      }
      __syncthreads();
    }
    last = ri[0];
    __syncthreads();
  }
}

// ---------------------------------------------------------------------------
// 2) Ball query: one wave32 per center; ballot-based ordered compaction of the
//    first K in-radius indices; pad with first hit (or 0 if none).
// ---------------------------------------------------------------------------
__global__ __launch_bounds__(128) void ballq_kernel(const float* __restrict__ xyz,
                                                    const float* __restrict__ nxyz,
                                                    int* __restrict__ ballIdx) {
  const int lane = threadIdx.x & 31;
  const int gw = blockIdx.x * 4 + (threadIdx.x >> 5);  // (b,s) center id
  const int b = gw / Ss;
  const float* xb = xyz + (size_t)b * 3 * Nn;
  const float* c = nxyz + (size_t)gw * 3;
  const float cx = c[0], cy = c[1], cz = c[2];
  int* out = ballIdx + (size_t)gw * Kk;
  int cnt = 0, firstN = 0;
  bool found = false;
  for (int n0 = 0; n0 < Nn; n0 += 32) {
    const int n = n0 + lane;
    float dx = xb[n] - cx, dy = xb[Nn + n] - cy, dz = xb[2 * Nn + n] - cz;
    bool inside = (dx * dx + dy * dy + dz * dz) <= R2;
    unsigned long long m = __ballot(inside);
    if (!found && m) { firstN = n0 + (int)__ffsll((unsigned long long)m) - 1; found = true; }
    int pos = cnt + (int)__popcll(m & ((1ull << lane) - 1ull));
    if (inside && pos < Kk) out[pos] = n;
    cnt += (int)__popcll(m);
    if (cnt >= Kk) break;
  }
  const int have = cnt < Kk ? cnt : Kk;
  const int fill = found ? firstN : 0;
  for (int p = have + lane; p < Kk; p += 32) out[p] = fill;
}

// ---------------------------------------------------------------------------
// 3) Layer-0 GEMM with fused gather: feat[0:3]=rel-xyz, feat[3:67]=points,
//    padded cin 67->68; fp32 WMMA 16x16x4. 4 waves/block -> 64 columns.
// ---------------------------------------------------------------------------
__global__ __launch_bounds__(128) void gemm0_kernel(const float* __restrict__ xyz,
                                                    const float* __restrict__ points,
                                                    const float* __restrict__ w0,
                                                    const int* __restrict__ ballIdx,
                                                    const float* __restrict__ nxyz,
                                                    float* __restrict__ out) {
  const int tid = threadIdx.x;
  const int lane = tid & 31, wave = tid >> 5;
  const int colBase = blockIdx.x * 64;
  const int b = colBase >> 15;  // / (S*K)
  __shared__ float W[64][68];
  __shared__ float X[68][68];
  for (int e = tid; e < 64 * 68; e += 128) {
    int rr = e / 68, cc = e % 68;
    W[rr][cc] = (cc < 67) ? w0[rr * 67 + cc] : 0.0f;
  }
  {
    const int col = tid & 63, half = tid >> 6;
    const int gc = colBase + col;
    const int s = (gc >> 5) & (Ss - 1);
    const int idx = ballIdx[gc];
    const float* xb = xyz + (size_t)b * 3 * Nn;
    const float* cen = nxyz + ((size_t)b * Ss + s) * 3;
    const float* pb = points + (size_t)b * CINc * Nn;
    const int c0 = half * 34;
    for (int c = c0; c < c0 + 34; ++c) {
      float v;
      if (c == 0)       v = xb[idx] - cen[0];
      else if (c == 1)  v = xb[Nn + idx] - cen[1];
      else if (c == 2)  v = xb[2 * Nn + idx] - cen[2];
      else if (c < 67)  v = pb[(c - 3) * Nn + idx];
      else              v = 0.0f;
      X[c][col] = v;
    }
  }
  __syncthreads();
  const int l15 = lane & 15, hi = lane >> 4;
  const int xcol = wave * 16 + l15;
  for (int mt = 0; mt < 4; ++mt) {
    v8f acc = {};
    for (int kk = 0; kk < 17; ++kk) {
      const int kb = kk * 4 + hi * 2;
      v2f av; av.x = W[mt * 16 + l15][kb]; av.y = W[mt * 16 + l15][kb + 1];
      v2f bv; bv.x = X[kb][xcol];          bv.y = X[kb + 1][xcol];
      acc = __builtin_amdgcn_wmma_f32_16x16x4_f32(false, av, false, bv,
                                                  (short)0, acc, false, false);
    }
    for (int r = 0; r < 8; ++r)
      out[(size_t)(mt * 16 + hi * 8 + r) * Mtot + colBase + xcol] = acc[r];
  }
}

// ---------------------------------------------------------------------------
// 4) Mid/late GEMM: applies previous layer's folded BN affine + relu on load,
//    then fp32 WMMA. Layer 1 runs in place (block reads its columns fully
//    before writing the same addresses). OC = 64 or 128.
// ---------------------------------------------------------------------------
__global__ __launch_bounds__(128) void gemm_mid_kernel(const float* __restrict__ in,
                                                       float* __restrict__ out,
                                                       const float* __restrict__ W,
                                                       int OC,
                                                       const float* __restrict__ bn) {
  __shared__ float Wl[128][64];
  __shared__ float X[64][68];
  const int tid = threadIdx.x, lane = tid & 31, wave = tid >> 5;
  const int colBase = blockIdx.x * 64;
  for (int e = tid; e < OC * 64; e += 128) Wl[e >> 6][e & 63] = W[e];
  for (int e = tid; e < 64 * 64; e += 128) {
    const int c = e >> 6, col = e & 63;
    const float a = bn[c], d = bn[128 + c];
    const float x = in[(size_t)c * Mtot + colBase + col];
    X[c][col] = fmaxf(a * x + d, 0.0f);
  }
  __syncthreads();
  const int l15 = lane & 15, hi = lane >> 4;
  const int xcol = wave * 16 + l15;
  const int MT = OC >> 4;
  for (int mt = 0; mt < MT; ++mt) {
    v8f acc = {};
    for (int kk = 0; kk < 16; ++kk) {
      const int kb = kk * 4 + hi * 2;
      v2f av; av.x = Wl[mt * 16 + l15][kb]; av.y = Wl[mt * 16 + l15][kb + 1];
      v2f bv; bv.x = X[kb][xcol];           bv.y = X[kb + 1][xcol];
      acc = __builtin_amdgcn_wmma_f32_16x16x4_f32(false, av, false, bv,
                                                  (short)0, acc, false, false);
    }
    for (int r = 0; r < 8; ++r)
      out[(size_t)(mt * 16 + hi * 8 + r) * Mtot + colBase + xcol] = acc[r];
  }
}

// ---------------------------------------------------------------------------
// 5) Deterministic per-channel sum / sumsq (one block per channel).
// ---------------------------------------------------------------------------
__global__ __launch_bounds__(256) void stats_kernel(const float* __restrict__ raw,
                                                    float* __restrict__ stats) {
  const int ch = blockIdx.x, tid = threadIdx.x;
  __shared__ float s1[256], s2[256];
  float a = 0.0f, b = 0.0f;
  const float* p = raw + (size_t)ch * Mtot;
  for (int i = tid; i < Mtot; i += 256) { float v = p[i]; a += v; b += v * v; }
  s1[tid] = a; s2[tid] = b;
  __syncthreads();
  for (int st = 128; st > 0; st >>= 1) {
    if (tid < st) { s1[tid] += s1[tid + st]; s2[tid] += s2[tid + st]; }
    __syncthreads();
  }
  if (tid == 0) { stats[ch] = s1[0]; stats[128 + ch] = s2[0]; }
}

// Fold (mu, var, g, b) into affine: y = a*x + d.
__global__ void finalize_kernel(const float* __restrict__ stats,
                                const float* __restrict__ g,
                                const float* __restrict__ bsh,
                                float* __restrict__ bn, int OC) {
  const int c = threadIdx.x;
  if (c < OC) {
    const float inv = 1.0f / (float)Mtot;
    const float mu = stats[c] * inv;
    const float var = stats[128 + c] * inv - mu * mu;
    const float a = g[c] * rsqrtf(var + 1e-5f);
    bn[c] = a;
    bn[128 + c] = bsh[c] - mu * a;
  }
}

// ---------------------------------------------------------------------------
// 6) Final: BN+relu on layer-2 raw, max over K, write new_points.
// ---------------------------------------------------------------------------
__global__ __launch_bounds__(256) void maxk_kernel(const float* __restrict__ raw2,
                                                   const float* __restrict__ bn,
                                                   float* __restrict__ out) {
  const int gid = blockIdx.x * 256 + threadIdx.x;  // B*128*S threads
  const int s = gid & (Ss - 1);
  const int c = (gid >> 10) & 127;
  const int b = gid >> 17;
  const float a = bn[c], d = bn[128 + c];
  const float* p = raw2 + (size_t)c * Mtot + ((size_t)(b * Ss + s)) * Kk;
  float m = 0.0f;  // relu output is >= 0
  for (int k = 0; k < Kk; ++k) m = fmaxf(m, fmaxf(a * p[k] + d, 0.0f));
  out[(size_t)Bb * 3 * Ss + ((size_t)b * 128 + c) * Ss + s] = m;
}

// ---------------------------------------------------------------------------
extern "C" void kernel_launch(void* const* d_in, const int* in_sizes, int n_in,
                              void* d_out, int out_size, void* d_ws, size_t ws_size,
                              hipStream_t stream) {
  (void)in_sizes; (void)n_in; (void)out_size; (void)ws_size;
  const float* xyz    = (const float*)d_in[0];
  const float* points = (const float*)d_in[1];
  const float* w0 = (const float*)d_in[2];
  const float* g0 = (const float*)d_in[3];
  const float* b0 = (const float*)d_in[4];
  const float* w1 = (const float*)d_in[5];
  const float* g1 = (const float*)d_in[6];
  const float* b1 = (const float*)d_in[7];
  const float* w2 = (const float*)d_in[8];
  const float* g2 = (const float*)d_in[9];
  const float* b2 = (const float*)d_in[10];
  float* out = (float*)d_out;

  char* ws = (char*)d_ws;
  size_t off = 0;
  int*   fpsIdx  = (int*)(ws + off);   off += (size_t)Bb * Ss * sizeof(int);
  int*   ballIdx = (int*)(ws + off);   off += (size_t)Bb * Ss * Kk * sizeof(int);
  float* nxyz    = (float*)(ws + off); off += (size_t)Bb * Ss * 3 * sizeof(float);
  float* stats   = (float*)(ws + off); off += 256 * sizeof(float);
  float* bn0     = (float*)(ws + off); off += 256 * sizeof(float);
  float* bn1     = (float*)(ws + off); off += 256 * sizeof(float);
  float* bn2     = (float*)(ws + off); off += 256 * sizeof(float);
  off = (off + 255) & ~(size_t)255;
  float* raw0    = (float*)(ws + off); off += (size_t)64 * Mtot * sizeof(float);
  float* raw2    = (float*)(ws + off); off += (size_t)128 * Mtot * sizeof(float);

  fps_kernel<<<Bb, 512, 0, stream>>>(xyz, fpsIdx, nxyz, out);
  ballq_kernel<<<(Bb * Ss) / 4, 128, 0, stream>>>(xyz, nxyz, ballIdx);

  gemm0_kernel<<<Mtot / 64, 128, 0, stream>>>(xyz, points, w0, ballIdx, nxyz, raw0);
  stats_kernel<<<64, 256, 0, stream>>>(raw0, stats);
  finalize_kernel<<<1, 128, 0, stream>>>(stats, g0, b0, bn0, 64);

  gemm_mid_kernel<<<Mtot / 64, 128, 0, stream>>>(raw0, raw0, w1, 64, bn0);  // in place
  stats_kernel<<<64, 256, 0, stream>>>(raw0, stats);
  finalize_kernel<<<1, 128, 0, stream>>>(stats, g1, b1, bn1, 64);

  gemm_mid_kernel<<<Mtot / 64, 128, 0, stream>>>(raw0, raw2, w2, 128, bn1);
  stats_kernel<<<128, 256, 0, stream>>>(raw2, stats);
  finalize_kernel<<<1, 128, 0, stream>>>(stats, g2, b2, bn2, 128);

  maxk_kernel<<<(Bb * 128 * Ss) / 256, 256, 0, stream>>>(raw2, bn2, out);
}